// CompiledSimulation_88046829568702
// MI455X (gfx1250) — compile-verified
//
#include <hip/hip_runtime.h>
#include <hip/hip_bf16.h>

typedef __attribute__((ext_vector_type(2))) float v2f;
typedef __attribute__((ext_vector_type(8))) float v8f;

#define EPSF    1e-9f
#define EXTENTF 12.0f
#define MAXC 32
#define MAXB 16

// ---------------------------------------------------------------- zero output
__global__ void zero_f32(float* __restrict__ p, int n) {
    int i = blockIdx.x * blockDim.x + threadIdx.x;
    if (i < n) p[i] = 0.0f;
}

// ------------------------------------------------- WMMA transform pre-pass
// tp[m,n,i] = sum_j R[m,i,j]*p[m,n,j] + pos[m,i] ; tn[m,n,i] = sum_j R[m,i,j]*nrm[m,n,j]
// One 16x16x4 fp32 WMMA per 16 vectors:
//   A (16x4): rows 0..2 = [ R | pos ], rows 3..15 = 0
//   B (4x16): columns = 16 input vectors [x,y,z,w]^T, w=1 for points (enables
//             the translation row), w=0 for normals.
//   D (16x16): rows 0..2 = transformed coords of the 16 vectors.
// fp32 A layout (ISA 7.12.2): VGPR0 = K0 (lanes 0-15) / K2 (lanes 16-31),
//                             VGPR1 = K1 / K3.  B mirrors this for its K rows.
// fp32 C/D layout: lane holds column N = lane&15; VGPR r = row r (lanes<16).
__global__ __launch_bounds__(256) void xform_wmma(
    const float* __restrict__ pts, const float* __restrict__ nrms,
    const float* __restrict__ pos, const float* __restrict__ rot,
    float* __restrict__ tp, float* __restrict__ tn, int M, int P)
{
    const int lane = threadIdx.x & 31;
    const int wave = (blockIdx.x * blockDim.x + threadIdx.x) >> 5;
    const int tilesPerM = P >> 4;
    if (wave >= M * tilesPerM) return;           // wave-uniform, EXEC stays full
    const int m    = wave / tilesPerM;
    const int base = m * P + (wave - m * tilesPerM) * 16;
    const int row  = lane & 15;
    const bool hi  = lane >= 16;

    v2f A;
    {
        float a0 = 0.0f, a1 = 0.0f;
        if (row < 3) {
            a0 = rot[m*9 + row*3 + (hi ? 2 : 0)];                 // K0 / K2
            a1 = hi ? pos[m*3 + row] : rot[m*9 + row*3 + 1];      // K1 / K3(=pos)
        }
        A.x = a0; A.y = a1;
    }

    const float* pp = pts  + (size_t)(base + row) * 3;
    const float* pn = nrms + (size_t)(base + row) * 3;
    v2f Bp, Bn;
    Bp.x = hi ? pp[2] : pp[0];   // K0 = x, K2 = z
    Bp.y = hi ? 1.0f  : pp[1];   // K1 = y, K3 = 1 (translation on)
    Bn.x = hi ? pn[2] : pn[0];
    Bn.y = hi ? 0.0f  : pn[1];   // K3 = 0 (translation off)

    v8f z = {};
    v8f Dp = __builtin_amdgcn_wmma_f32_16x16x4_f32(false, A, false, Bp, (short)0, z, false, false);
    v8f Dn = __builtin_amdgcn_wmma_f32_16x16x4_f32(false, A, false, Bn, (short)0, z, false, false);

    if (!hi) {  // lanes 0-15 hold rows 0..7 -> rows 0..2 are the coords of point N=lane
        float* op = tp + (size_t)(base + row) * 3;
        float* on = tn + (size_t)(base + row) * 3;
        op[0] = Dp[0]; op[1] = Dp[1]; op[2] = Dp[2];
        on[0] = Dn[0]; on[1] = Dn[1]; on[2] = Dn[2];
    }
}

// ---------------------------------------------------------------- ray kernel
__global__ __launch_bounds__(256) void ray_trace(
    const float* __restrict__ tp,  const float* __restrict__ tn,
    const float* __restrict__ pts, const float* __restrict__ nrms,
    const float* __restrict__ pos, const float* __restrict__ rot,
    const float* __restrict__ sources,
    const float* __restrict__ cp1, const float* __restrict__ cp2,
    const float* __restrict__ crad,
    const float* __restrict__ bp1, const float* __restrict__ bp2,
    const float* __restrict__ spp, const float* __restrict__ spn,
    const int* __restrict__ pH, const int* __restrict__ pW,
    float* __restrict__ out,
    int M, int S, int P, int NC, int NB, int useTp, long long total)
{
    __shared__ float sAx[MAXC], sAy[MAXC], sAz[MAXC], sLen[MAXC], sR2[MAXC];
    __shared__ float sPx[MAXC], sPy[MAXC], sPz[MAXC];
    __shared__ float sLo[MAXB][3], sHiB[MAXB][3];

    const int tid = threadIdx.x;
    if (tid < NC) {
        float ax = cp2[tid*3+0] - cp1[tid*3+0];
        float ay = cp2[tid*3+1] - cp1[tid*3+1];
        float az = cp2[tid*3+2] - cp1[tid*3+2];
        float L  = sqrtf(ax*ax + ay*ay + az*az);
        float iv = 1.0f / (L + EPSF);
        sAx[tid] = ax*iv; sAy[tid] = ay*iv; sAz[tid] = az*iv;
        sLen[tid] = L;
        float r = crad[tid]; sR2[tid] = r*r;
        sPx[tid] = cp1[tid*3+0]; sPy[tid] = cp1[tid*3+1]; sPz[tid] = cp1[tid*3+2];
    } else if (tid >= 64 && tid < 64 + NB) {
        int b = tid - 64;
        sLo[b][0]=bp1[b*3+0]; sLo[b][1]=bp1[b*3+1]; sLo[b][2]=bp1[b*3+2];
        sHiB[b][0]=bp2[b*3+0]; sHiB[b][1]=bp2[b*3+1]; sHiB[b][2]=bp2[b*3+2];
    }
    __syncthreads();

    long long gid = (long long)blockIdx.x * blockDim.x + tid;
    if (gid >= total) return;
    int p = (int)(gid % P);
    long long rr = gid / P;
    int s = (int)(rr % S);
    int m = (int)(rr / S);

    float ox, oy, oz, nx, ny, nz;
    if (useTp) {
        const float* o3 = tp + (size_t)(m*P + p)*3;
        ox=o3[0]; oy=o3[1]; oz=o3[2];
        const float* n3 = tn + (size_t)(m*P + p)*3;
        nx=n3[0]; ny=n3[1]; nz=n3[2];
    } else {  // fallback if d_ws is too small: scalar transform per ray
        const float* R = rot + (size_t)m*9;
        const float* q = pts  + (size_t)(m*P + p)*3;
        const float* w = nrms + (size_t)(m*P + p)*3;
        ox = R[0]*q[0]+R[1]*q[1]+R[2]*q[2] + pos[m*3+0];
        oy = R[3]*q[0]+R[4]*q[1]+R[5]*q[2] + pos[m*3+1];
        oz = R[6]*q[0]+R[7]*q[1]+R[8]*q[2] + pos[m*3+2];
        nx = R[0]*w[0]+R[1]*w[1]+R[2]*w[2];
        ny = R[3]*w[0]+R[4]*w[1]+R[5]*w[2];
        nz = R[6]*w[0]+R[7]*w[1]+R[8]*w[2];
    }

    float sxr = sources[s*3+0], syr = sources[s*3+1], szr = sources[s*3+2];
    float dx = ox - sxr, dy = oy - syr, dz = oz - szr;
    float invl = 1.0f / sqrtf(dx*dx + dy*dy + dz*dz);
    dx *= invl; dy *= invl; dz *= invl;            // d (source -> mirror)
    float ux = -dx, uy = -dy, uz = -dz;            // occlusion ray dir = -d

    bool hit = false;
    for (int c = 0; c < NC; ++c) {
        float ax = sAx[c], ay = sAy[c], az = sAz[c];
        float ocx = ox - sPx[c], ocy = oy - sPy[c], ocz = oz - sPz[c];
        float ua = ux*ax + uy*ay + uz*az;
        float oa = ocx*ax + ocy*ay + ocz*az;
        float ddx = ux - ua*ax, ddy = uy - ua*ay, ddz = uz - ua*az;
        float mx = ocx - oa*ax, my = ocy - oa*ay, mz = ocz - oa*az;
        float Aq = ddx*ddx + ddy*ddy + ddz*ddz;
        float Bq = 2.0f * (mx*ddx + my*ddy + mz*ddz);
        float Cq = mx*mx + my*my + mz*mz - sR2[c];
        float disc = Bq*Bq - 4.0f*Aq*Cq;
        if (disc > 0.0f) {
            float sq  = sqrtf(disc);
            float i2a = 1.0f / (2.0f*Aq + EPSF);
            float t1 = (-Bq - sq) * i2a;
            float t2 = (-Bq + sq) * i2a;
            float x1 = oa + t1*ua;
            float x2 = oa + t2*ua;
            float L  = sLen[c];
            bool h1 = (t1 > EPSF) && (x1 >= 0.0f) && (x1 <= L);
            bool h2 = (t2 > EPSF) && (x2 >= 0.0f) && (x2 <= L);
            if (h1 || h2) { hit = true; break; }
        }
    }
    if (!hit) {
        float iux = 1.0f / ((fabsf(ux) < EPSF) ? EPSF : ux);
        float iuy = 1.0f / ((fabsf(uy) < EPSF) ? EPSF : uy);
        float iuz = 1.0f / ((fabsf(uz) < EPSF) ? EPSF : uz);
        for (int b = 0; b < NB; ++b) {
            float t0x = (sLo[b][0]-ox)*iux, t1x = (sHiB[b][0]-ox)*iux;
            float t0y = (sLo[b][1]-oy)*iuy, t1y = (sHiB[b][1]-oy)*iuy;
            float t0z = (sLo[b][2]-oz)*iuz, t1z = (sHiB[b][2]-oz)*iuz;
            float tmin = fmaxf(fmaxf(fminf(t0x,t1x), fminf(t0y,t1y)), fminf(t0z,t1z));
            float tmax = fminf(fminf(fmaxf(t0x,t1x), fmaxf(t0y,t1y)), fmaxf(t0z,t1z));
            if (tmax >= fmaxf(tmin, EPSF)) { hit = true; break; }
        }
    }
    if (hit) return;   // shadow == 0 -> contributes nothing

    float dn = dx*nx + dy*ny + dz*nz;
    float rx = dx - 2.0f*dn*nx;
    float ry = dy - 2.0f*dn*ny;
    float rz = dz - 2.0f*dn*nz;
    float cosv = fabsf(dn);
    float snx = spn[0], sny = spn[1], snz = spn[2];
    float denom = rx*snx + ry*sny + rz*snz;
    float tnum  = (spp[0]-ox)*snx + (spp[1]-oy)*sny + (spp[2]-oz)*snz;
    float t = tnum / (denom + EPSF);
    float qx = ox + t*rx;
    float qy = oy + t*ry;
    int W = *pW, H = *pH;
    float fx = (qx + EXTENTF) / (2.0f*EXTENTF) * (float)W;
    float fy = (qy + EXTENTF) / (2.0f*EXTENTF) * (float)H;
    int ix = (int)floorf(fx);
    int iy = (int)floorf(fy);
    if (ix >= 0 && ix < W && iy >= 0 && iy < H) {
        atomicAdd(&out[(size_t)iy*W + ix], cosv);   // hw global_atomic_add_f32, L2-resident
    }
}

// ---------------------------------------------------------------- launch
extern "C" void kernel_launch(void* const* d_in, const int* in_sizes, int n_in,
                              void* d_out, int out_size, void* d_ws, size_t ws_size,
                              hipStream_t stream) {
    const float* sources = (const float*)d_in[0];
    const float* pts     = (const float*)d_in[1];
    const float* nrms    = (const float*)d_in[2];
    const float* pos     = (const float*)d_in[3];
    const float* rot     = (const float*)d_in[4];
    const float* cp1     = (const float*)d_in[5];
    const float* cp2     = (const float*)d_in[6];
    const float* crad    = (const float*)d_in[7];
    const float* bp1     = (const float*)d_in[8];
    const float* bp2     = (const float*)d_in[9];
    const float* spp     = (const float*)d_in[10];
    const float* spn     = (const float*)d_in[11];
    const int*   pH      = (const int*)d_in[12];
    const int*   pW      = (const int*)d_in[13];

    int S  = in_sizes[0] / 3;
    int M  = in_sizes[3] / 3;
    int P  = in_sizes[1] / (3 * M);
    int NC = in_sizes[7];     if (NC > MAXC) NC = MAXC;
    int NB = in_sizes[8] / 3; if (NB > MAXB) NB = MAXB;

    float* out = (float*)d_out;
    float* tp  = (float*)d_ws;
    float* tn  = tp + (size_t)M * P * 3;

    size_t need = (size_t)M * P * 6 * sizeof(float);
    int useTp = (ws_size >= need && (P % 16) == 0) ? 1 : 0;

    zero_f32<<<(out_size + 255) / 256, 256, 0, stream>>>(out, out_size);

    if (useTp) {
        int tiles = M * (P / 16);
        int tb = (tiles + 7) / 8;          // 8 wave32 per 256-thread block
        xform_wmma<<<tb, 256, 0, stream>>>(pts, nrms, pos, rot, tp, tn, M, P);
    }

    long long total = (long long)M * S * P;
    long long nb = (total + 255) / 256;
    ray_trace<<<dim3((unsigned int)nb), 256, 0, stream>>>(
        tp, tn, pts, nrms, pos, rot, sources, cp1, cp2, crad, bp1, bp2,
        spp, spn, pH, pW, out, M, S, P, NC, NB, useTp, total);
}